// TargetOpinionPairRepresentation_35553739276882
// MI455X (gfx1250) — compile-verified
//
#include <hip/hip_runtime.h>
#include <cstdint>
#include <cstddef>

// ---------------------------------------------------------------------------
// TargetOpinionPairRepresentation: pure gather + concat (store-bandwidth bound)
//   out[b, t*O+o, :] = concat(spans[b, tgt[b,t]], spans[b, opn[b,o]],
//                             dist_emb[bucket(span_idx[tgt], span_idx[opn])])
// ~218 MB of stores @ 23.3 TB/s => ~9.5us floor. No matmul -> no WMMA.
// Data plane uses the CDNA5 async global<->LDS DMA path (ASYNCcnt):
//   global_load_async_to_lds_b128 -> s_wait_asynccnt 0
//     -> global_store_async_from_lds_b128 (TH=NT: output has zero reuse,
//        keep L2 capacity for the 64x-reused gathered span rows)
// ---------------------------------------------------------------------------

namespace {
constexpr int Bc = 8, Sc = 4096, Dc = 768, Tc = 64, Oc = 64, Ec = 128;
constexpr int ROW  = 2 * Dc + Ec;   // 1664 floats per output row
constexpr int ROW4 = ROW / 4;       // 416 float4 chunks per row (= blockDim)
constexpr int D4   = Dc / 4;        // 192 float4 per span row
}  // namespace

#define LDS_AS __attribute__((address_space(3)))
#define GLB_AS __attribute__((address_space(1)))

typedef int v4i __attribute__((ext_vector_type(4)));

#if defined(__HIP_DEVICE_COMPILE__) &&                                        \
    __has_builtin(__builtin_amdgcn_global_load_async_to_lds_b128) &&          \
    __has_builtin(__builtin_amdgcn_global_store_async_from_lds_b128)
#define USE_ASYNC_LDS 1
#else
#define USE_ASYNC_LDS 0
#endif

// gfx1250 CPol: TH in low bits; 0 = RT (default), 1 = NT (non-temporal).
// Temporal hints affect cache placement only, never correctness.
#define CPOL_RT 0
#define CPOL_NT 1

__global__ __launch_bounds__(ROW4) void topair_gather_kernel(
    const float* __restrict__ spans,          // [B, S, D]
    const float* __restrict__ dist_emb,       // [14, E]
    const int*   __restrict__ span_idx,       // [S, 2]
    const int*   __restrict__ tgt_idx,        // [B, T]
    const int*   __restrict__ opn_idx,        // [B, O]
    float*       __restrict__ out)            // [B, T*O, ROW]
{
    // ---- block-uniform index math (lowers to SALU / scalar loads) ----
    const int r    = blockIdx.x;        // row in [0, B*T*O)
    const int b    = r >> 12;           // / (T*O) == 4096
    const int pair = r & 4095;
    const int t    = pair >> 6;         // repeat(target, O): t-major
    const int o    = pair & 63;         // tile(opinion, T)

    const int ti = tgt_idx[b * Tc + t];
    const int oi = opn_idx[b * Oc + o];

    const int a0 = span_idx[2 * ti + 0];
    const int a1 = span_idx[2 * ti + 1];
    const int c0 = span_idx[2 * oi + 0];
    const int c1 = span_idx[2 * oi + 1];
    int u = a1 - c0; u = (u < 0) ? -u : u;   // |b - c|
    int v = a0 - c1; v = (v < 0) ? -v : v;   // |a - d|
    const int dist = (u < v) ? u : v;

    // bucket = searchsorted(bins, dist, 'right') - 1 = (#bins <= dist) - 1
    const int bins[14] = {0, 1, 2, 3, 4, 5, 7, 8, 15, 16, 31, 32, 63, 64};
    int bucket = -1;
#pragma unroll
    for (int k = 0; k < 14; ++k) bucket += (dist >= bins[k]) ? 1 : 0;

    // ---- per-lane 16B source/dest ----
    const int i = threadIdx.x;  // 0..415
    const float* src;
    if (i < D4) {
        src = spans + ((size_t)b * Sc + (size_t)ti) * Dc + 4 * i;
    } else if (i < 2 * D4) {
        src = spans + ((size_t)b * Sc + (size_t)oi) * Dc + 4 * (i - D4);
    } else {
        src = dist_emb + (size_t)bucket * Ec + 4 * (i - 2 * D4);
    }
    float* dst = out + (size_t)r * ROW + 4 * i;

#if USE_ASYNC_LDS
    // Stage through LDS via the CDNA5 async DMA path: data never touches
    // VGPRs. Each lane loads and stores its own 16B slot, so a per-wave
    // ASYNCcnt wait fully orders the LDS write before the LDS read; no
    // workgroup barrier needed. s_endpgm's implicit wait-idle drains the
    // trailing async store.
    __shared__ float stage[ROW];                        // 6656 B per block
    LDS_AS char* lbase = (LDS_AS char*)stage;           // addrspacecast -> LDS
    LDS_AS v4i*  lslot = (LDS_AS v4i*)(lbase + 16 * i);
    GLB_AS v4i*  gsrc  = (GLB_AS v4i*)(const_cast<float*>(src));
    GLB_AS v4i*  gdst  = (GLB_AS v4i*)dst;

    __builtin_amdgcn_global_load_async_to_lds_b128(gsrc, lslot, 0, CPOL_RT);
#if __has_builtin(__builtin_amdgcn_s_wait_asynccnt)
    __builtin_amdgcn_s_wait_asynccnt(0);
#else
    asm volatile("s_wait_asynccnt 0" ::: "memory");
#endif
    __builtin_amdgcn_global_store_async_from_lds_b128(gdst, lslot, 0, CPOL_NT);
#else
    // Fallback: plain wide gather (global_load_b128 / global_store_b128).
    *(float4*)dst = *(const float4*)src;
#endif
}

extern "C" void kernel_launch(void* const* d_in, const int* in_sizes, int n_in,
                              void* d_out, int out_size, void* d_ws, size_t ws_size,
                              hipStream_t stream) {
    const float* spans     = (const float*)d_in[0];   // [8, 4096, 768] f32
    const float* dist_emb  = (const float*)d_in[1];   // [14, 128] f32
    const int*   span_idx  = (const int*)d_in[2];     // [4096, 2] i32
    const int*   tgt_idx   = (const int*)d_in[3];     // [8, 64] i32
    const int*   opn_idx   = (const int*)d_in[4];     // [8, 64] i32
    float*       out       = (float*)d_out;           // [8, 4096, 1664] f32

    (void)in_sizes; (void)n_in; (void)out_size; (void)d_ws; (void)ws_size;

    const dim3 grid(Bc * Tc * Oc);   // 32768 rows
    const dim3 block(ROW4);          // 416 threads = 13 wave32, 1 float4/lane
    topair_gather_kernel<<<grid, block, 0, stream>>>(
        spans, dist_emb, span_idx, tgt_idx, opn_idx, out);
}